// SelfSupervisedLoss_41042707481154
// MI455X (gfx1250) — compile-verified
//
#include <hip/hip_runtime.h>
#include <math.h>

typedef __attribute__((ext_vector_type(2))) float v2f;
typedef __attribute__((ext_vector_type(8))) float v8f;

#define ZETA_C   0.005f
#define ALPHA_C  0.5f
#define NB_C     8
#define INTERVAL_C 0.1f

// ---------- helpers ----------------------------------------------------------

// monotonic float->uint mapping so atomicMin(uint) implements float min
__device__ __forceinline__ unsigned fenc(float f) {
  unsigned u = __float_as_uint(f);
  return (u & 0x80000000u) ? ~u : (u | 0x80000000u);
}
__device__ __forceinline__ float fdec(unsigned u) {
  return (u & 0x80000000u) ? __uint_as_float(u & 0x7FFFFFFFu)
                           : __uint_as_float(~u);
}
__device__ __forceinline__ bool lexlt(float da, int ia, float db, int ib) {
  return (da < db) || ((da == db) && (ia < ib));
}
// fast exp: single v_mul + v_exp_f32 (TRANS32, co-executes with VALU)
__device__ __forceinline__ float fexp(float x) { return __expf(x); }

// ---------- init workspace ---------------------------------------------------

__global__ void k_init(unsigned* colMin, float* colSum, float* sb, float* nb,
                       float* scSum, float* rdSum, int BN, int B) {
  int t = blockIdx.x * blockDim.x + threadIdx.x;
  if (t < BN) { colMin[t] = 0xFFFFFFFFu; colSum[t] = 0.0f; }
  if (t < B)  { sb[t] = 0.0f; nb[t] = 0.0f; }
  if (t == 0) { scSum[0] = 0.0f; rdSum[0] = 0.0f; }
}

// ---------- pack (B,3,N) -> (B,N,4) + squared norms --------------------------

__global__ void k_pack(const float* __restrict__ pc1, const float* __restrict__ pc2,
                       const float* __restrict__ prf,
                       float* P1, float* P2, float* PW, float* FL,
                       float* n1, float* n2, float* nw, int B, int N) {
  int t = blockIdx.x * blockDim.x + threadIdx.x;
  int BN = B * N;
  if (t >= BN) return;
  int b = t / N, n = t - b * N;
  size_t base = (size_t)(b * 3) * N + n;
  float x1 = pc1[base], y1 = pc1[base + N], z1 = pc1[base + 2 * (size_t)N];
  float x2 = pc2[base], y2 = pc2[base + N], z2 = pc2[base + 2 * (size_t)N];
  float fx = prf[base], fy = prf[base + N], fz = prf[base + 2 * (size_t)N];
  float wx = x1 + fx, wy = y1 + fy, wz = z1 + fz;
  P1[4 * t + 0] = x1; P1[4 * t + 1] = y1; P1[4 * t + 2] = z1; P1[4 * t + 3] = 0.0f;
  P2[4 * t + 0] = x2; P2[4 * t + 1] = y2; P2[4 * t + 2] = z2; P2[4 * t + 3] = 0.0f;
  PW[4 * t + 0] = wx; PW[4 * t + 1] = wy; PW[4 * t + 2] = wz; PW[4 * t + 3] = 0.0f;
  FL[4 * t + 0] = fx; FL[4 * t + 1] = fy; FL[4 * t + 2] = fz; FL[4 * t + 3] = 0.0f;
  n1[t] = x1 * x1 + y1 * y1 + z1 * z1;
  n2[t] = x2 * x2 + y2 * y2 + z2 * z2;
  nw[t] = wx * wx + wy * wy + wz * wz;
}

// ---------- chamfer + density: WMMA f32 16x16x4 dot products -----------------
// One wave per 16-row strip of one batch; streams 16-col tiles of pc2.
// Two WMMAs per tile: warped (min-reduction) and unwarped (density exp-sums).

__global__ __launch_bounds__(32)
void k_chamfer(const float* __restrict__ P1, const float* __restrict__ P2,
               const float* __restrict__ PW,
               const float* __restrict__ n1, const float* __restrict__ n2,
               const float* __restrict__ nw,
               float* rowMin, float* rowSum,
               unsigned* colMin, float* colSum, int N) {
  const int bpB = N >> 4;
  const int b   = blockIdx.x / bpB;
  const int i0  = (blockIdx.x - b * bpB) << 4;
  const int lane = threadIdx.x;
  const int g  = lane >> 4;     // half-wave group selects K pair / M half
  const int lr = lane & 15;
  const size_t bo = (size_t)b * N;

  // A fragments (16x4, K=3 padded): lane lr holds row i0+lr, K = {2g, 2g+1}
  const int arow = i0 + lr;
  v2f aW, a1;
  aW.x = PW[(bo + arow) * 4 + 2 * g]; aW.y = PW[(bo + arow) * 4 + 2 * g + 1];
  a1.x = P1[(bo + arow) * 4 + 2 * g]; a1.y = P1[(bo + arow) * 4 + 2 * g + 1];

  // C-fragment row for register r on this lane: i0 + r + 8*g
  float nwv[8], n1v[8], rmin[8], rsum[8];
#pragma unroll
  for (int r = 0; r < 8; ++r) {
    int i = i0 + r + 8 * g;
    nwv[r] = nw[bo + i];
    n1v[r] = n1[bo + i];
    rmin[r] = 3.0e38f;
    rsum[r] = 0.0f;
  }

  for (int j0 = 0; j0 < N; j0 += 16) {
    const int j = j0 + lr;
    // B fragment (4x16): lane lr holds col j0+lr, K = {2g, 2g+1}
    v2f bv;
    bv.x = P2[(bo + j) * 4 + 2 * g];
    bv.y = P2[(bo + j) * 4 + 2 * g + 1];
    const float n2j = n2[bo + j];

    v8f cw = {};
    v8f c1 = {};
    cw = __builtin_amdgcn_wmma_f32_16x16x4_f32(false, aW, false, bv,
                                               (short)0, cw, false, false);
    c1 = __builtin_amdgcn_wmma_f32_16x16x4_f32(false, a1, false, bv,
                                               (short)0, c1, false, false);

    float ccmin = 3.0e38f, ccsum = 0.0f;
#pragma unroll
    for (int r = 0; r < 8; ++r) {
      float dW = nwv[r] + n2j - 2.0f * cw[r];      // |pc1w_i - pc2_j|^2
      rmin[r] = fminf(rmin[r], dW);
      ccmin   = fminf(ccmin, dW);
      float d1 = n1v[r] + n2j - 2.0f * c1[r];      // |pc1_i - pc2_j|^2
      float e  = fexp(-0.5f * d1);                 // bandwidth = 1
      rsum[r] += e;
      ccsum   += e;
    }
    // fold the two M-halves (lanes l <-> l+16 share the same column j)
    ccmin = fminf(ccmin, __shfl_xor(ccmin, 16));
    ccsum = ccsum + __shfl_xor(ccsum, 16);
    if (lane < 16) {
      atomicMin(&colMin[bo + j], fenc(ccmin));
      atomicAdd(&colSum[bo + j], ccsum);
    }
  }

  // reduce row quantities across the 16 columns (lanes within each half-wave)
#pragma unroll
  for (int m = 1; m < 16; m <<= 1) {
#pragma unroll
    for (int r = 0; r < 8; ++r) {
      rmin[r] = fminf(rmin[r], __shfl_xor(rmin[r], m));
      rsum[r] += __shfl_xor(rsum[r], m);
    }
  }
  if (lr == 0) {          // lane 0 -> rows i0..i0+7, lane 16 -> rows i0+8..i0+15
#pragma unroll
    for (int r = 0; r < 8; ++r) {
      int i = i0 + r + 8 * g;
      rowMin[bo + i] = rmin[r];
      rowSum[bo + i] = rsum[r];
    }
  }
}

// ---------- chamfer finalize: masks + relu + global sum ----------------------

__global__ void k_chamfin(const float* __restrict__ rowMin,
                          const float* __restrict__ rowSum,
                          const unsigned* __restrict__ colMin,
                          const float* __restrict__ colSum,
                          float* scSum, int BN, int N) {
  int t = blockIdx.x * blockDim.x + threadIdx.x;
  float v = 0.0f;
  if (t < BN) {
    float inv = 1.0f / (2.5f * (float)N);
    float m1 = (rowSum[t] * inv > ZETA_C) ? 1.0f : 0.0f;
    float m2 = (colSum[t] * inv > ZETA_C) ? 1.0f : 0.0f;
    float v1 = fmaxf(rowMin[t] - 0.01f, 0.0f) * m1;
    float v2 = fmaxf(fdec(colMin[t]) - 0.01f, 0.0f) * m2;
    v = v1 + v2;
  }
#pragma unroll
  for (int m = 16; m > 0; m >>= 1) v += __shfl_xor(v, m);
  if ((threadIdx.x & 31) == 0) atomicAdd(scSum, v);
}

// ---------- kNN + smoothness: one wave per point -----------------------------

__global__ __launch_bounds__(32)
void k_knn(const float* __restrict__ P1, const float* __restrict__ n1,
           const float* __restrict__ FL, float* sb, float* nb, int N) {
  const int b = blockIdx.x / N;
  const int i = blockIdx.x - b * N;
  const int lane = threadIdx.x;
  const size_t bo = (size_t)b * N;

  const float pix = P1[(bo + i) * 4 + 0];
  const float piy = P1[(bo + i) * 4 + 1];
  const float piz = P1[(bo + i) * 4 + 2];
  const float ni  = n1[bo + i];

  float d9[9]; int i9[9];
#pragma unroll
  for (int k = 0; k < 9; ++k) { d9[k] = 3.0e38f; i9[k] = 0x7FFFFFFF; }

  for (int j = lane; j < N; j += 32) {
    float qx = P1[(bo + j) * 4 + 0];
    float qy = P1[(bo + j) * 4 + 1];
    float qz = P1[(bo + j) * 4 + 2];
    float dot = pix * qx + piy * qy + piz * qz;
    float d = ni + n1[bo + j] - 2.0f * dot;
    if (lexlt(d, j, d9[8], i9[8])) {
      d9[8] = d; i9[8] = j;
#pragma unroll
      for (int t = 8; t > 0; --t) {
        if (lexlt(d9[t], i9[t], d9[t - 1], i9[t - 1])) {
          float td = d9[t]; d9[t] = d9[t - 1]; d9[t - 1] = td;
          int   ti = i9[t]; i9[t] = i9[t - 1]; i9[t - 1] = ti;
        }
      }
    }
  }

  // log2(32) rounds of symmetric sorted-list merge across lanes
  for (int m = 1; m < 32; m <<= 1) {
    float od[9]; int oi[9];
#pragma unroll
    for (int k = 0; k < 9; ++k) {
      od[k] = __shfl_xor(d9[k], m);
      oi[k] = __shfl_xor(i9[k], m);
    }
    float ndv[9]; int niv[9];
    int a = 0, c = 0;
#pragma unroll
    for (int k = 0; k < 9; ++k) {
      bool ta = lexlt(d9[a], i9[a], od[c], oi[c]);
      ndv[k] = ta ? d9[a] : od[c];
      niv[k] = ta ? i9[a] : oi[c];
      if (ta) ++a; else ++c;
    }
#pragma unroll
    for (int k = 0; k < 9; ++k) { d9[k] = ndv[k]; i9[k] = niv[k]; }
  }

  if (lane == 0) {
    // entry 0 is self (distance ~0): drop it, keep 8 neighbors
    float fix = FL[(bo + i) * 4 + 0];
    float fiy = FL[(bo + i) * 4 + 1];
    float fiz = FL[(bo + i) * 4 + 2];
    float se = 0.0f, sn = 0.0f;
#pragma unroll
    for (int k = 1; k <= NB_C; ++k) {
      int j = i9[k];
      float dk = fmaxf(d9[k], 0.0f);
      float z  = fexp(-dk / ALPHA_C);   // exp(-d/alpha)
      float e  = fexp(z);               // softmax numerator exp(z), z in (0,1]
      float dx = FL[(bo + j) * 4 + 0] - fix;
      float dy = FL[(bo + j) * 4 + 1] - fiy;
      float dz = FL[(bo + j) * 4 + 2] - fiz;
      float diff = sqrtf(dx * dx + dy * dy + dz * dz);
      se += e;
      sn += e * diff;
    }
    atomicAdd(&sb[b], se);
    atomicAdd(&nb[b], sn);
  }
}

// ---------- radial displacement ---------------------------------------------

__global__ void k_rd(const float* __restrict__ pc1, const float* __restrict__ prf,
                     const float* __restrict__ vel, float* rdSum, int B, int N) {
  int t = blockIdx.x * blockDim.x + threadIdx.x;
  int BN = B * N;
  float v = 0.0f;
  if (t < BN) {
    int b = t / N, n = t - b * N;
    size_t base = (size_t)(b * 3) * N + n;
    float x = pc1[base], y = pc1[base + N], z = pc1[base + 2 * (size_t)N];
    float fx = prf[base], fy = prf[base + N], fz = prf[base + 2 * (size_t)N];
    float fr = (fx * x + fy * y + fz * z) * __frsqrt_rn(x * x + y * y + z * z);
    v = fabsf(vel[t] * INTERVAL_C - fr);
  }
#pragma unroll
  for (int m = 16; m > 0; m >>= 1) v += __shfl_xor(v, m);
  if ((threadIdx.x & 31) == 0) atomicAdd(rdSum, v);
}

// ---------- final combine ----------------------------------------------------

__global__ void k_final(const float* scSum, const float* rdSum,
                        const float* sb, const float* nb,
                        float* out, int B, int N) {
  if (blockIdx.x == 0 && threadIdx.x == 0) {
    float invBN = 1.0f / (float)(B * N);
    float sc = scSum[0] * invBN;
    float ss = 0.0f;
    for (int b = 0; b < B; ++b) ss += nb[b] / sb[b];
    ss /= (float)B;
    float rd = rdSum[0] * invBN;
    out[0] = sc + ss + rd;
    out[1] = sc;
    out[2] = ss;
    out[3] = rd;
  }
}

// ---------- host launcher ----------------------------------------------------

extern "C" void kernel_launch(void* const* d_in, const int* in_sizes, int n_in,
                              void* d_out, int out_size, void* d_ws, size_t ws_size,
                              hipStream_t stream) {
  const float* pc1 = (const float*)d_in[0];
  const float* pc2 = (const float*)d_in[1];
  const float* prf = (const float*)d_in[2];
  const float* vel = (const float*)d_in[3];
  float* out = (float*)d_out;

  const int B  = 4;
  const int N  = in_sizes[3] / B;   // vel1 is (B, N)
  const int BN = B * N;

  // workspace layout (floats)
  float*    P1     = (float*)d_ws;
  float*    P2     = P1 + 4 * (size_t)BN;
  float*    PW     = P2 + 4 * (size_t)BN;
  float*    FL     = PW + 4 * (size_t)BN;
  float*    n1     = FL + 4 * (size_t)BN;
  float*    n2     = n1 + BN;
  float*    nw     = n2 + BN;
  float*    rowMin = nw + BN;
  float*    rowSum = rowMin + BN;
  unsigned* colMin = (unsigned*)(rowSum + BN);
  float*    colSum = (float*)(colMin + BN);
  float*    sb     = colSum + BN;
  float*    nb     = sb + B;
  float*    scSum  = nb + B;
  float*    rdSum  = scSum + 1;

  const int T = 256;
  k_init   <<<(BN + T - 1) / T, T, 0, stream>>>(colMin, colSum, sb, nb, scSum, rdSum, BN, B);
  k_pack   <<<(BN + T - 1) / T, T, 0, stream>>>(pc1, pc2, prf, P1, P2, PW, FL, n1, n2, nw, B, N);
  k_chamfer<<<B * (N / 16), 32, 0, stream>>>(P1, P2, PW, n1, n2, nw, rowMin, rowSum, colMin, colSum, N);
  k_chamfin<<<(BN + T - 1) / T, T, 0, stream>>>(rowMin, rowSum, colMin, colSum, scSum, BN, N);
  k_knn    <<<BN, 32, 0, stream>>>(P1, n1, FL, sb, nb, N);
  k_rd     <<<(BN + T - 1) / T, T, 0, stream>>>(pc1, prf, vel, rdSum, B, N);
  k_final  <<<1, 32, 0, stream>>>(scSum, rdSum, sb, nb, out, B, N);
}